// CanonicalEncoder_15857019256955
// MI455X (gfx1250) — compile-verified
//
#include <hip/hip_runtime.h>
#include <math.h>

typedef float v2f __attribute__((ext_vector_type(2)));
typedef float v8f __attribute__((ext_vector_type(8)));

#define B_   8
#define N_   4096
#define KNN  16
#define MFPS 1024
#define EPSN 1e-12f

// ---------------------------------------------------------------------------
// KNN via V_WMMA_F32_16X16X4_F32, candidates on M-axis, queries on N-axis.
// The K=3 slot carries |p|^2 (A side) and -0.5 (B side), so the WMMA directly
// produces c = q.p - 0.5|p|^2. Since d = |q|^2 - 2c and |q|^2 is constant per
// lane (one query per lane/column), nearest-neighbor selection == top-K by
// LARGEST c straight out of the accumulator: one VALU compare per candidate,
// zero LDS and zero extra arithmetic in the inner loop. Lane l and lane l+16
// keep two sorted half-lists (candidate rows 0-7 vs 8-15 of each tile) for
// the same query, merged once at the end.
// ---------------------------------------------------------------------------
__global__ void knn_wmma_kernel(const float* __restrict__ query,
                                const float* __restrict__ pts,
                                int* __restrict__ knn_idx,
                                int Q, int N)
{
  const int b    = blockIdx.y;
  const int q0   = blockIdx.x * 16;
  const int lane = threadIdx.x;
  const int half = lane >> 4;       // 0: candidate rows 0-7, 1: rows 8-15
  const int l15  = lane & 15;
  const float* qb = query + (size_t)b * Q * 3;
  const float* pb = pts   + (size_t)b * N * 3;

  // B-matrix (4x16): query N=l15. lanes 0-15 hold (qx,qy) K=0,1;
  // lanes 16-31 hold (qz,-0.5) K=2,3. Loop-invariant.
  const int   qm = q0 + l15;
  const float qx = qb[qm*3+0], qy = qb[qm*3+1], qz = qb[qm*3+2];
  v2f bmat;
  if (half == 0) { bmat.x = qx; bmat.y = qy; }
  else           { bmat.x = qz; bmat.y = -0.5f; }

  // Descending top-16 by score c = q.p - 0.5|p|^2  (max c == min distance).
  float bs[KNN]; int bidx[KNN];
  #pragma unroll
  for (int j = 0; j < KNN; ++j) { bs[j] = -3.0e38f; bidx[j] = 0; }

  const int mbase = half * 8;

  #pragma unroll 2
  for (int t = 0; t < N; t += 16) {
    // A-matrix (16x4): candidate M=l15 holds (px,py) / (pz,|p|^2).
    const int   pm = t + l15;
    const float px = pb[pm*3+0], py = pb[pm*3+1], pz = pb[pm*3+2];
    const float pp = px*px + py*py + pz*pz;
    v2f amat;
    if (half == 0) { amat.x = px; amat.y = py; }
    else           { amat.x = pz; amat.y = pp; }

    v8f c = {0.f,0.f,0.f,0.f,0.f,0.f,0.f,0.f};
    c = __builtin_amdgcn_wmma_f32_16x16x4_f32(false, amat, false, bmat,
                                              (short)0, c, false, false);

    // Lane holds c[m][n] for n = its query, m = mbase + r.
    #pragma unroll
    for (int r = 0; r < 8; ++r) {
      const float s  = c[r];
      const int   gi = t + mbase + r;
      if (s > bs[KNN-1]) {
        bs[KNN-1] = s; bidx[KNN-1] = gi;
        #pragma unroll
        for (int j = KNN-1; j > 0; --j) {
          if (bs[j] > bs[j-1]) {
            float ts = bs[j]; bs[j] = bs[j-1]; bs[j-1] = ts;
            int   ti = bidx[j]; bidx[j] = bidx[j-1]; bidx[j-1] = ti;
          }
        }
      }
    }
  }

  // Merge the two sorted (descending) half-lists -> 16 best (largest c).
  __shared__ float ds_[32][KNN + 1];
  __shared__ int   is_[32][KNN + 1];
  #pragma unroll
  for (int j = 0; j < KNN; ++j) { ds_[lane][j] = bs[j]; is_[lane][j] = bidx[j]; }
  __syncthreads();
  if (lane < 16) {
    int ia = 0, ib = 0;
    int* outp = knn_idx + ((size_t)b * Q + q0 + lane) * KNN;
    #pragma unroll
    for (int j = 0; j < KNN; ++j) {
      const float da = ds_[lane][ia];
      const float db = ds_[lane + 16][ib];
      if (da >= db) { outp[j] = is_[lane][ia];      ++ia; }
      else          { outp[j] = is_[lane + 16][ib]; ++ib; }
    }
  }
}

// ---------------------------------------------------------------------------
// Per-point normal estimation: 16-neighbor mean, 3x3 covariance, smallest
// eigenvector via closed-form (Cardano) + cross-product eigenvector.
// ---------------------------------------------------------------------------
__global__ void normals_kernel(const float* __restrict__ xyz,
                               const int* __restrict__ idx,
                               float* __restrict__ normals,
                               float* __restrict__ local_mean)
{
  const int g = blockIdx.x * blockDim.x + threadIdx.x;
  if (g >= B_ * N_) return;
  const int b = g / N_;
  const float* xb = xyz + (size_t)b * N_ * 3;
  const int*   ip = idx + (size_t)g * KNN;

  float nx[KNN], ny[KNN], nz[KNN];
  float mx = 0.f, my = 0.f, mz = 0.f;
  #pragma unroll
  for (int j = 0; j < KNN; ++j) {
    const int t = ip[j];
    nx[j] = xb[t*3+0]; ny[j] = xb[t*3+1]; nz[j] = xb[t*3+2];
    mx += nx[j]; my += ny[j]; mz += nz[j];
  }
  mx *= (1.f/KNN); my *= (1.f/KNN); mz *= (1.f/KNN);
  local_mean[(size_t)g*3+0] = mx;
  local_mean[(size_t)g*3+1] = my;
  local_mean[(size_t)g*3+2] = mz;

  float a=0.f, bb=0.f, cc=0.f, dxy=0.f, exz=0.f, fyz=0.f;
  #pragma unroll
  for (int j = 0; j < KNN; ++j) {
    const float cx = nx[j]-mx, cy = ny[j]-my, cz = nz[j]-mz;
    a += cx*cx; bb += cy*cy; cc += cz*cz;
    dxy += cx*cy; exz += cx*cz; fyz += cy*cz;
  }
  const float s = 1.f / (KNN - 1);
  a *= s; bb *= s; cc *= s; dxy *= s; exz *= s; fyz *= s;

  float vx = 1.f, vy = 0.f, vz = 0.f;
  const float p1 = dxy*dxy + exz*exz + fyz*fyz;
  const float qv = (a + bb + cc) * (1.f/3.f);
  const float aq = a - qv, bq = bb - qv, cq = cc - qv;
  const float p2 = aq*aq + bq*bq + cq*cq + 2.f*p1;
  if (p2 > 1e-24f) {
    const float p  = sqrtf(p2 * (1.f/6.f));
    const float iv = 1.f / p;
    const float b00 = aq*iv, b11 = bq*iv, b22 = cq*iv;
    const float b01 = dxy*iv, b02 = exz*iv, b12 = fyz*iv;
    float r = 0.5f * (b00*(b11*b22 - b12*b12)
                    - b01*(b01*b22 - b12*b02)
                    + b02*(b01*b12 - b11*b02));
    r = fminf(1.f, fmaxf(-1.f, r));
    const float phi  = acosf(r) * (1.f/3.f);
    const float lmin = qv + 2.f*p*cosf(phi + 2.0943951023931953f); // +2pi/3
    // eigenvector: largest cross product of rows of (A - lmin I)
    const float r0x=a-lmin, r0y=dxy,     r0z=exz;
    const float r1x=dxy,    r1y=bb-lmin, r1z=fyz;
    const float r2x=exz,    r2y=fyz,     r2z=cc-lmin;
    const float c0x=r0y*r1z-r0z*r1y, c0y=r0z*r1x-r0x*r1z, c0z=r0x*r1y-r0y*r1x;
    const float c1x=r0y*r2z-r0z*r2y, c1y=r0z*r2x-r0x*r2z, c1z=r0x*r2y-r0y*r2x;
    const float c2x=r1y*r2z-r1z*r2y, c2y=r1z*r2x-r1x*r2z, c2z=r1x*r2y-r1y*r2x;
    const float l0=c0x*c0x+c0y*c0y+c0z*c0z;
    const float l1=c1x*c1x+c1y*c1y+c1z*c1z;
    const float l2=c2x*c2x+c2y*c2y+c2z*c2z;
    float bl = l0; vx=c0x; vy=c0y; vz=c0z;
    if (l1 > bl) { bl=l1; vx=c1x; vy=c1y; vz=c1z; }
    if (l2 > bl) { bl=l2; vx=c2x; vy=c2y; vz=c2z; }
    if (bl <= 1e-30f) { vx=1.f; vy=0.f; vz=0.f; }
  }
  const float nn = fmaxf(sqrtf(vx*vx+vy*vy+vz*vz), EPSN);
  normals[(size_t)g*3+0] = vx / nn;
  normals[(size_t)g*3+1] = vy / nn;
  normals[(size_t)g*3+2] = vz / nn;
}

// ---------------------------------------------------------------------------
// Neighbor-averaged normal, normal-projected displacement removal.
// ---------------------------------------------------------------------------
__global__ void update_kernel(const float* __restrict__ xyz,
                              const int* __restrict__ idx,
                              const float* __restrict__ normals,
                              const float* __restrict__ local_mean,
                              float* __restrict__ mean_normal,
                              float* __restrict__ xyz_nu)
{
  const int g = blockIdx.x * blockDim.x + threadIdx.x;
  if (g >= B_ * N_) return;
  const int b = g / N_;
  const float* nb = normals + (size_t)b * N_ * 3;
  const int*   ip = idx + (size_t)g * KNN;

  float sx = 0.f, sy = 0.f, sz = 0.f;
  #pragma unroll
  for (int j = 0; j < KNN; ++j) {
    const int t = ip[j];
    sx += nb[t*3+0]; sy += nb[t*3+1]; sz += nb[t*3+2];
  }
  sx *= (1.f/KNN); sy *= (1.f/KNN); sz *= (1.f/KNN);
  const float nn = fmaxf(sqrtf(sx*sx+sy*sy+sz*sz), EPSN);
  sx /= nn; sy /= nn; sz /= nn;
  mean_normal[(size_t)g*3+0] = sx;
  mean_normal[(size_t)g*3+1] = sy;
  mean_normal[(size_t)g*3+2] = sz;

  const float x = xyz[(size_t)g*3+0], y = xyz[(size_t)g*3+1], z = xyz[(size_t)g*3+2];
  const float dx = x - local_mean[(size_t)g*3+0];
  const float dy = y - local_mean[(size_t)g*3+1];
  const float dz = z - local_mean[(size_t)g*3+2];
  const float t  = sx*dx + sy*dy + sz*dz;
  xyz_nu[(size_t)g*3+0] = x - sx*t;
  xyz_nu[(size_t)g*3+1] = y - sy*t;
  xyz_nu[(size_t)g*3+2] = z - sz*t;
}

// ---------------------------------------------------------------------------
// Farthest point sampling: one block per batch, 256 threads, all 4096 points
// register-resident (16/thread). Sequential 1024-step loop; argmax via wave32
// shuffles + one small LDS round (3 barriers/step). Ties -> lowest index.
// ---------------------------------------------------------------------------
__global__ void fps_kernel(const float* __restrict__ xyz_nu,
                           int* __restrict__ fps_idx,
                           float* __restrict__ xyz_fps)
{
  const int b   = blockIdx.x;
  const int tid = threadIdx.x;
  const float* pb = xyz_nu + (size_t)b * N_ * 3;

  __shared__ int   cur_s;
  __shared__ float lx_s, ly_s, lz_s;
  __shared__ float wd_s[8];
  __shared__ int   wi_s[8];
  __shared__ int   fidx_s[MFPS];

  float px[16], py[16], pz[16], md[16];
  #pragma unroll
  for (int i = 0; i < 16; ++i) {
    const int n = i * 256 + tid;
    px[i] = pb[n*3+0]; py[i] = pb[n*3+1]; pz[i] = pb[n*3+2];
    md[i] = 3.0e38f;
  }
  if (tid == 0) cur_s = 0;
  __syncthreads();

  for (int k = 0; k < MFPS; ++k) {
    if (tid == 0) {
      const int cur = cur_s;
      fidx_s[k] = cur;
      lx_s = pb[cur*3+0]; ly_s = pb[cur*3+1]; lz_s = pb[cur*3+2];
    }
    __syncthreads();
    const float lx = lx_s, ly = ly_s, lz = lz_s;

    float best = -1.0f; int bi = 0;
    #pragma unroll
    for (int i = 0; i < 16; ++i) {
      const float dx = px[i]-lx, dy = py[i]-ly, dz = pz[i]-lz;
      const float d  = dx*dx + dy*dy + dz*dz;
      md[i] = fminf(md[i], d);
      const int n = i * 256 + tid;
      if (md[i] > best || (md[i] == best && n < bi)) { best = md[i]; bi = n; }
    }
    #pragma unroll
    for (int off = 16; off > 0; off >>= 1) {
      const float od = __shfl_down(best, off, 32);
      const int   oi = __shfl_down(bi,   off, 32);
      if (od > best || (od == best && oi < bi)) { best = od; bi = oi; }
    }
    if ((tid & 31) == 0) { wd_s[tid >> 5] = best; wi_s[tid >> 5] = bi; }
    __syncthreads();
    if (tid == 0) {
      float bB = wd_s[0]; int bI = wi_s[0];
      for (int w = 1; w < 8; ++w)
        if (wd_s[w] > bB || (wd_s[w] == bB && wi_s[w] < bI)) { bB = wd_s[w]; bI = wi_s[w]; }
      cur_s = bI;
    }
    __syncthreads();
  }

  for (int k = tid; k < MFPS; k += 256) {
    const int sidx = fidx_s[k];
    fps_idx[b * MFPS + k] = sidx;
    xyz_fps[((size_t)b * MFPS + k)*3 + 0] = pb[sidx*3+0];
    xyz_fps[((size_t)b * MFPS + k)*3 + 1] = pb[sidx*3+1];
    xyz_fps[((size_t)b * MFPS + k)*3 + 2] = pb[sidx*3+2];
  }
}

// ---------------------------------------------------------------------------
// Tangential smoothing: out = local_mean_fps + n (n . (xyz_fps - local_mean))
// ---------------------------------------------------------------------------
__global__ void final_kernel(const float* __restrict__ xyz_fps,
                             const float* __restrict__ xyz_nu,
                             const int* __restrict__ idx2,
                             const float* __restrict__ mean_normal,
                             float* __restrict__ out)
{
  const int g = blockIdx.x * blockDim.x + threadIdx.x;
  if (g >= B_ * MFPS) return;
  const int b = g / MFPS;
  const float* pb = xyz_nu      + (size_t)b * N_ * 3;
  const float* nb = mean_normal + (size_t)b * N_ * 3;
  const int*   ip = idx2 + (size_t)g * KNN;

  float mx=0.f,my=0.f,mz=0.f, sx=0.f,sy=0.f,sz=0.f;
  #pragma unroll
  for (int j = 0; j < KNN; ++j) {
    const int t = ip[j];
    mx += pb[t*3+0]; my += pb[t*3+1]; mz += pb[t*3+2];
    sx += nb[t*3+0]; sy += nb[t*3+1]; sz += nb[t*3+2];
  }
  mx *= (1.f/KNN); my *= (1.f/KNN); mz *= (1.f/KNN);
  sx *= (1.f/KNN); sy *= (1.f/KNN); sz *= (1.f/KNN);
  const float nn = fmaxf(sqrtf(sx*sx+sy*sy+sz*sz), EPSN);
  sx /= nn; sy /= nn; sz /= nn;

  const float fx = xyz_fps[(size_t)g*3+0];
  const float fy = xyz_fps[(size_t)g*3+1];
  const float fz = xyz_fps[(size_t)g*3+2];
  const float t  = sx*(fx-mx) + sy*(fy-my) + sz*(fz-mz);
  out[(size_t)g*3+0] = mx + sx*t;
  out[(size_t)g*3+1] = my + sy*t;
  out[(size_t)g*3+2] = mz + sz*t;
}

// ---------------------------------------------------------------------------
extern "C" void kernel_launch(void* const* d_in, const int* in_sizes, int n_in,
                              void* d_out, int out_size, void* d_ws, size_t ws_size,
                              hipStream_t stream) {
  (void)in_sizes; (void)n_in; (void)out_size; (void)ws_size;
  const float* xyz = (const float*)d_in[0];

  char* ws = (char*)d_ws;
  size_t off = 0;
  auto carve = [&](size_t bytes) -> void* {
    void* p = ws + off;
    off = (off + bytes + 255) & ~(size_t)255;
    return p;
  };
  int*   idx1        = (int*)  carve((size_t)B_ * N_ * KNN * sizeof(int));
  float* local_mean  = (float*)carve((size_t)B_ * N_ * 3 * sizeof(float));
  float* normals     = (float*)carve((size_t)B_ * N_ * 3 * sizeof(float));
  float* mean_normal = (float*)carve((size_t)B_ * N_ * 3 * sizeof(float));
  float* xyz_nu      = (float*)carve((size_t)B_ * N_ * 3 * sizeof(float));
  int*   fps_idx     = (int*)  carve((size_t)B_ * MFPS * sizeof(int));
  float* xyz_fps     = (float*)carve((size_t)B_ * MFPS * 3 * sizeof(float));
  int*   idx2        = (int*)  carve((size_t)B_ * MFPS * KNN * sizeof(int));

  // 1) self-KNN (WMMA computes full distance score; VALU only selects)
  knn_wmma_kernel<<<dim3(N_/16, B_), 32, 0, stream>>>(xyz, xyz, idx1, N_, N_);
  // 2) covariance normals + local means
  normals_kernel<<<(B_*N_ + 255)/256, 256, 0, stream>>>(xyz, idx1, normals, local_mean);
  // 3) neighbor-averaged normals + normal-projected update
  update_kernel<<<(B_*N_ + 255)/256, 256, 0, stream>>>(xyz, idx1, normals, local_mean,
                                                       mean_normal, xyz_nu);
  // 4) farthest point sampling (one block per batch)
  fps_kernel<<<B_, 256, 0, stream>>>(xyz_nu, fps_idx, xyz_fps);
  // 5) KNN of FPS samples against updated cloud (WMMA again)
  knn_wmma_kernel<<<dim3(MFPS/16, B_), 32, 0, stream>>>(xyz_fps, xyz_nu, idx2, MFPS, N_);
  // 6) tangential smoothing -> output [B, 1024, 3]
  final_kernel<<<(B_*MFPS + 255)/256, 256, 0, stream>>>(xyz_fps, xyz_nu, idx2,
                                                        mean_normal, (float*)d_out);
}